// rnnosc3ode_70308614636170
// MI455X (gfx1250) — compile-verified
//
#include <hip/hip_runtime.h>
#include <hip/hip_bf16.h>

typedef float v2f __attribute__((ext_vector_type(2)));
typedef float v8f __attribute__((ext_vector_type(8)));

constexpr int B   = 1024;
constexpr int T   = 2048;
constexpr int NIN = 8;
constexpr int NHI = 64;
constexpr int NE  = 3 * (T - 1) + 1;   // 6142 eval points
constexpr int NEP = 3 * T;             // padded rows in ws
constexpr float L2E = 1.4426950408889634f;

__device__ __forceinline__ float fast_sigmoid(float p) {
    // 1/(1+exp(-p)) via v_exp_f32 (2^x) + v_rcp_f32
    return __builtin_amdgcn_rcpf(1.0f + __builtin_amdgcn_exp2f(-p * L2E));
}
__device__ __forceinline__ float fast_expneg(float a) {   // exp(-a)
    return __builtin_amdgcn_exp2f(-a * L2E);
}
__device__ __forceinline__ float tanh_fast(float x) {
    float r;
    // CDNA5 TRANS op; pad with v_nops so the TRANS result/source hazard is
    // satisfied inside the asm block (compiler can't schedule around it).
    asm("v_tanh_f32 %0, %1\n\tv_nop\n\tv_nop" : "=v"(r) : "v"(x));
    return r;
}

// ---------------------------------------------------------------------------
// Kernel 1: invscale[e][b] = exp(-(tau + dparam(t_e))) for all eval points.
// One wave per (b, 16-grid-point tile). P = u@Wni^T + bni via f32 WMMA
// (16x16x4, K=8 as two chained WMMAs, 4 N-tiles). P tile -> LDS, then each
// lane evaluates lerp+sigmoid+dot for its eval points.
// ---------------------------------------------------------------------------
__global__ __launch_bounds__(32)
void k1_dparam(const float* __restrict__ u, const float* __restrict__ Wni,
               const float* __restrict__ bni, const float* __restrict__ Wli,
               const float* __restrict__ bli, const float* __restrict__ tau,
               float* __restrict__ invsc)
{
    const int lane = threadIdx.x;
    const int tt   = blockIdx.x;          // 0..T/16-1
    const int b    = blockIdx.y;          // 0..B-1
    const int t0   = tt * 16;
    const int mc   = lane & 15;           // A-row / D-col index
    const int hf   = lane >> 4;           // wave half

    __shared__ float P[17][65];           // rows t0..t0+16, 64 cols (+pad)
    __shared__ float WliS[64];

    WliS[lane]      = Wli[lane];
    WliS[lane + 32] = Wli[lane + 32];

    // A operands: A[m][k] = u[b][t0+m][k], lane half picks k = {2h, 2h+1}
    const float* up = u + ((size_t)b * T + t0 + mc) * NIN + 2 * hf;
    v2f a0 = *(const v2f*)(up);           // k-step 0 (k = 0..3)
    v2f a1 = *(const v2f*)(up + 4);       // k-step 1 (k = 4..7)

#pragma unroll
    for (int nt = 0; nt < 4; ++nt) {
        int n = nt * 16 + mc;
        const float* wp = Wni + n * NIN + 2 * hf;   // B[k][n] = Wni[n][k]
        v2f b0 = *(const v2f*)(wp);
        v2f b1 = *(const v2f*)(wp + 4);
        float bias = bni[n];
        v8f c = {bias, bias, bias, bias, bias, bias, bias, bias};
        c = __builtin_amdgcn_wmma_f32_16x16x4_f32(false, a0, false, b0,
                                                  (short)0, c, false, false);
        c = __builtin_amdgcn_wmma_f32_16x16x4_f32(false, a1, false, b1,
                                                  (short)0, c, false, false);
        // D: lane holds col n for rows m = v + 8*half
#pragma unroll
        for (int v = 0; v < 8; ++v) P[v + 8 * hf][n] = c[v];
    }

    // Extra grid row t0+16 (needed by last interval's interpolations)
    if (t0 + 16 < T) {
        const float* ux = u + ((size_t)b * T + t0 + 16) * NIN;
        float uv[8];
#pragma unroll
        for (int k = 0; k < 8; ++k) uv[k] = ux[k];
#pragma unroll
        for (int rep = 0; rep < 2; ++rep) {
            int n = lane + 32 * rep;
            float acc = bni[n];
#pragma unroll
            for (int k = 0; k < 8; ++k) acc = fmaf(uv[k], Wni[n * NIN + k], acc);
            P[16][n] = acc;
        }
    }
    __syncthreads();

    const float blis = bli[0];
    const float taus = tau[0];
    // 48 eval points per tile: e = 3*t0 + j, j = 3*il + r
    for (int j = lane; j < 48; j += 32) {
        int e = 3 * t0 + j;
        if (e >= NE) continue;
        int il = j / 3, r = j - 3 * il;
        int ih = il + (r != 0);
        float frac = (r == 0) ? 0.0f : (r == 1 ? (1.0f / 3.0f) : (2.0f / 3.0f));
        float acc = 0.0f;
        for (int n = 0; n < 64; ++n) {
            float plo = P[il][n];
            float phi = P[ih][n];
            float p = fmaf(phi - plo, frac, plo);
            acc = fmaf(WliS[n], fast_sigmoid(p), acc);
        }
        invsc[(size_t)e * B + b] = fast_expneg(taus + acc + blis);
    }
}

// ---------------------------------------------------------------------------
// Kernel 2: sequential 3/8-rule RK4 scan. One lane per batch chain.
// Writes state planes xsp[k][t][b] (coalesced across lanes).
// ---------------------------------------------------------------------------
__global__ __launch_bounds__(32)
void k2_scan(const float* __restrict__ invsc, float* __restrict__ xsp)
{
    const int b = blockIdx.x * 32 + threadIdx.x;
    const float W0c = -20.0f, W1c = 10.0f;
    float x0 = 0.4736f, x1 = 0.8745f, x2 = 1.8497f;

    float* xp0 = xsp;
    float* xp1 = xsp + (size_t)T * B;
    float* xp2 = xsp + 2 * (size_t)T * B;
    xp0[b] = x0; xp1[b] = x1; xp2[b] = x2;          // t = 0 row

    const float stopf = (float)(2e-11 * 2047.0 * 1.0e9);
    const float delta = stopf / 2047.0f;             // grid spacing

    float s0 = invsc[b];
    float s1 = invsc[(size_t)1 * B + b];
    float s2 = invsc[(size_t)2 * B + b];
    float s3 = invsc[(size_t)3 * B + b];
    float tprev = 0.0f;

    for (int i = 0; i < T - 1; ++i) {
        float n1 = 0.0f, n2 = 0.0f, n3 = 0.0f;
        if (i < T - 2) {                             // prefetch next step's scales
            const float* p = invsc + (size_t)(3 * i + 4) * B + b;
            n1 = p[0]; n2 = p[B]; n3 = p[2 * B];
        }
        float tnext = (float)(i + 1) * delta;
        float h = tnext - tprev;
        tprev = tnext;

        // k1 = F(x)*s0
        float ta = tanh_fast(x0), tb = tanh_fast(x1), tc = tanh_fast(x2);
        float k1x = (-x0 + tb) * s0;
        float k1y = (-x1 + tc) * s0;
        float k1z = (-x2 + W0c * ta + W1c * tb) * s0;
        // k2 = F(x + h*k1/3)*s1
        float h3 = h * (1.0f / 3.0f);
        float y0 = fmaf(h3, k1x, x0), y1 = fmaf(h3, k1y, x1), y2 = fmaf(h3, k1z, x2);
        float ua = tanh_fast(y0), ub = tanh_fast(y1), uc = tanh_fast(y2);
        float k2x = (-y0 + ub) * s1;
        float k2y = (-y1 + uc) * s1;
        float k2z = (-y2 + W0c * ua + W1c * ub) * s1;
        // k3 = F(x + h*(k2 - k1/3))*s2
        float z0 = fmaf(h, k2x - k1x * (1.0f / 3.0f), x0);
        float z1 = fmaf(h, k2y - k1y * (1.0f / 3.0f), x1);
        float z2 = fmaf(h, k2z - k1z * (1.0f / 3.0f), x2);
        float va = tanh_fast(z0), vb = tanh_fast(z1), vc = tanh_fast(z2);
        float k3x = (-z0 + vb) * s2;
        float k3y = (-z1 + vc) * s2;
        float k3z = (-z2 + W0c * va + W1c * vb) * s2;
        // k4 = F(x + h*(k1 - k2 + k3))*s3
        float w0 = fmaf(h, k1x - k2x + k3x, x0);
        float w1 = fmaf(h, k1y - k2y + k3y, x1);
        float w2 = fmaf(h, k1z - k2z + k3z, x2);
        float wa = tanh_fast(w0), wb = tanh_fast(w1), wc = tanh_fast(w2);
        float k4x = (-w0 + wb) * s3;
        float k4y = (-w1 + wc) * s3;
        float k4z = (-w2 + W0c * wa + W1c * wb) * s3;
        // x += h*(k1 + 3k2 + 3k3 + k4)/8
        float c8 = h * 0.125f;
        x0 = fmaf(c8, k1x + 3.0f * k2x + 3.0f * k3x + k4x, x0);
        x1 = fmaf(c8, k1y + 3.0f * k2y + 3.0f * k3y + k4y, x1);
        x2 = fmaf(c8, k1z + 3.0f * k2z + 3.0f * k3z + k4z, x2);

        size_t o = (size_t)(i + 1) * B + b;
        xp0[o] = x0; xp1[o] = x1; xp2[o] = x2;

        s0 = s3; s1 = n1; s2 = n2; s3 = n3;
    }
}

// ---------------------------------------------------------------------------
// Kernel 3: readout out[b][t] = Wl . sigmoid(x@Wn^T + bn) + bl via f32 WMMA
// (K=3 padded to 4). One wave per 16 flattened (t,b) rows.
// ---------------------------------------------------------------------------
__global__ __launch_bounds__(32)
void k3_readout(const float* __restrict__ xsp, const float* __restrict__ Wn,
                const float* __restrict__ bn, const float* __restrict__ Wl,
                const float* __restrict__ blv, float* __restrict__ out)
{
    const int lane = threadIdx.x;
    const int f0   = blockIdx.x * 16;     // flattened row base: f = t*B + b
    const int mc   = lane & 15;
    const int hf   = lane >> 4;
    const size_t TB = (size_t)T * B;

    // A[m][k] = xsp[k][f0+m], k<3, pad k=3 with 0; lane half h holds k={2h,2h+1}
    float a0e, a1e;
    if (hf == 0) { a0e = xsp[f0 + mc];          a1e = xsp[TB + f0 + mc]; }
    else         { a0e = xsp[2 * TB + f0 + mc]; a1e = 0.0f; }
    v2f A; A.x = a0e; A.y = a1e;

    float partial[8];
#pragma unroll
    for (int v = 0; v < 8; ++v) partial[v] = 0.0f;

#pragma unroll
    for (int nt = 0; nt < 4; ++nt) {
        int n = nt * 16 + mc;
        float b0e, b1e;                    // B[k][n] = Wn[n][k], pad k=3
        if (hf == 0) { b0e = Wn[n * 3 + 0]; b1e = Wn[n * 3 + 1]; }
        else         { b0e = Wn[n * 3 + 2]; b1e = 0.0f; }
        v2f Bv; Bv.x = b0e; Bv.y = b1e;
        float bias = bn[n];
        v8f c = {bias, bias, bias, bias, bias, bias, bias, bias};
        c = __builtin_amdgcn_wmma_f32_16x16x4_f32(false, A, false, Bv,
                                                  (short)0, c, false, false);
        float wl = Wl[n];
#pragma unroll
        for (int v = 0; v < 8; ++v)
            partial[v] = fmaf(fast_sigmoid(c[v]), wl, partial[v]);
    }

    __shared__ float red[32][9];
#pragma unroll
    for (int v = 0; v < 8; ++v) red[lane][v] = partial[v];
    __syncthreads();

    if (lane < 16) {                       // row m = lane; reduce its 16 cols
        int v = lane & 7, hh = lane >> 3;
        float acc = blv[0];
#pragma unroll
        for (int l = 0; l < 16; ++l) acc += red[16 * hh + l][v];
        int f  = f0 + lane;
        int t  = f >> 10;                  // f = t*B + b, B = 1024
        int bi = f & 1023;
        out[(size_t)bi * T + t] = acc;     // output layout (B, T, 1)
    }
}

extern "C" void kernel_launch(void* const* d_in, const int* in_sizes, int n_in,
                              void* d_out, int out_size, void* d_ws, size_t ws_size,
                              hipStream_t stream) {
    const float* u   = (const float*)d_in[0];
    const float* Wni = (const float*)d_in[1];
    const float* bni = (const float*)d_in[2];
    const float* Wli = (const float*)d_in[3];
    const float* bli = (const float*)d_in[4];
    const float* tau = (const float*)d_in[5];
    const float* Wn  = (const float*)d_in[6];
    const float* bn  = (const float*)d_in[7];
    const float* Wl  = (const float*)d_in[8];
    const float* bl  = (const float*)d_in[9];
    float* out   = (float*)d_out;
    float* invsc = (float*)d_ws;                       // NEP * B floats (~25 MB)
    float* xsp   = invsc + (size_t)NEP * B;            // 3 * T * B floats (~25 MB)

    hipLaunchKernelGGL(k1_dparam, dim3(T / 16, B), dim3(32), 0, stream,
                       u, Wni, bni, Wli, bli, tau, invsc);
    hipLaunchKernelGGL(k2_scan, dim3(B / 32), dim3(32), 0, stream, invsc, xsp);
    hipLaunchKernelGGL(k3_readout, dim3((B * T) / 16), dim3(32), 0, stream,
                       xsp, Wn, bn, Wl, bl, out);
}